// GCNQCritic_91233695301735
// MI455X (gfx1250) — compile-verified
//
#include <hip/hip_runtime.h>
#include <hip/hip_bf16.h>

typedef __attribute__((ext_vector_type(2))) float v2f;
typedef __attribute__((ext_vector_type(8))) float v8f;

#define STATE_DIM 128
#define HID 128

// ---------------------------------------------------------------------------
// Degree / normalization kernels
// ---------------------------------------------------------------------------
__global__ void deg_init_k(float* __restrict__ deg, int n) {
    int i = blockIdx.x * blockDim.x + threadIdx.x;
    if (i < n) deg[i] = 1.0f;  // self-loop
}

__global__ void deg_edge_k(const int* __restrict__ dst, float* __restrict__ deg, int e) {
    int i = blockIdx.x * blockDim.x + threadIdx.x;
    if (i < e) atomicAdd(&deg[dst[i]], 1.0f);
}

__global__ void dinv_k(const float* __restrict__ deg, float* __restrict__ dinv, int n) {
    int i = blockIdx.x * blockDim.x + threadIdx.x;
    if (i < n) dinv[i] = rsqrtf(deg[i]);
}

// ---------------------------------------------------------------------------
// agg[i] = xw[i] * dinv[i]^2   (self-loop term; fully initializes agg)
// 32 lanes per node, one float4 per lane
// ---------------------------------------------------------------------------
__global__ void selfloop_k(const float* __restrict__ xw, const float* __restrict__ dinv,
                           float* __restrict__ agg, int n) {
    long long t = (long long)blockIdx.x * blockDim.x + threadIdx.x;
    int i0 = (int)(t >> 5);
    if (i0 >= n) return;                       // wave-uniform exit
    int i = __builtin_amdgcn_readfirstlane(i0);
    int c = (threadIdx.x & 31) * 4;
    float w = dinv[i] * dinv[i];
    float4 v = *(const float4*)(xw + (size_t)i * HID + c);
    float4 o = make_float4(v.x * w, v.y * w, v.z * w, v.w * w);
    *(float4*)(agg + (size_t)i * HID + c) = o;
}

// ---------------------------------------------------------------------------
// agg[dst] += xw[src] * dinv[src]*dinv[dst]
// One wave per edge (t>>5 is wave-uniform) -> edge metadata forced to the
// scalar path via readfirstlane; lanes do the float4 gather + 4 f32 atomics.
// ---------------------------------------------------------------------------
__global__ void edge_scatter_k(const float* __restrict__ xw, const float* __restrict__ dinv,
                               const int* __restrict__ src, const int* __restrict__ dst,
                               float* __restrict__ agg, int e) {
    long long t = (long long)blockIdx.x * blockDim.x + threadIdx.x;
    int e0 = (int)(t >> 5);
    if (e0 >= e) return;                        // wave-uniform exit
    int ei = __builtin_amdgcn_readfirstlane(e0);
    int c = (threadIdx.x & 31) * 4;
    int s = src[ei], d = dst[ei];
    float w = dinv[s] * dinv[d];
    float4 v = *(const float4*)(xw + (size_t)s * HID + c);
    float* base = agg + (size_t)d * HID + c;
    atomicAdd(base + 0, v.x * w);
    atomicAdd(base + 1, v.y * w);
    atomicAdd(base + 2, v.z * w);
    atomicAdd(base + 3, v.w * w);
}

// ---------------------------------------------------------------------------
// Cooperative load of W[rows x 128] (row-major) into even/odd row-paired LDS:
// lds[ ((r>>1)*128 + c)*2 + (r&1) ] = W[r][c]
// so a B-fragment {W[kb][col], W[kb+1][col]} (kb even) is one aligned b64.
// ---------------------------------------------------------------------------
__device__ inline void load_w_paired(const float* __restrict__ Wg, float* lds,
                                     int rows, int tid, int nthreads) {
    int total = rows * 128;
    for (int i = tid; i < total; i += nthreads) {
        int r = i >> 7, c = i & 127;
        lds[(((r >> 1) << 7) + c) * 2 + (r & 1)] = Wg[i];
    }
}

// ---------------------------------------------------------------------------
// out[N,128] = f(A)[N,128] @ W[128,128];  f = relu(.+bias) iff HASBIAS.
// 256 threads = 8 waves; each wave computes a 16x128 strip with
// V_WMMA_F32_16X16X4_F32 (8 accumulators, 32-step K loop), branch-free body:
// OOB rows load clamped row n-1 (their C rows are OOB too and never stored).
// ---------------------------------------------------------------------------
template <bool HASBIAS>
__global__ __launch_bounds__(256) void wmma_gemm128_k(
    const float* __restrict__ A, const float* __restrict__ W,
    const float* __restrict__ bias,
    float* __restrict__ out, int n) {
    __shared__ float Wlds[128 * 128];
    __shared__ float Blds[128];
    int tid = threadIdx.x;
    load_w_paired(W, Wlds, 128, tid, 256);
    if (HASBIAS && tid < 128) Blds[tid] = bias[tid];
    __syncthreads();

    int lane = tid & 31;
    int wave = tid >> 5;
    int m  = lane & 15;   // row (A) / col (B,C) within tile
    int kg = lane >> 4;   // lanes 16-31: K+2,K+3 for A/B frags; rows +8 for C/D
    int rowBase = blockIdx.x * 128 + wave * 16;
    int row = rowBase + m;
    int rowC = row < n ? row : n - 1;           // clamp; no predication in loop
    const float* arow = A + (size_t)rowC * HID;

    v8f acc[8];
#pragma unroll
    for (int j = 0; j < 8; ++j) acc[j] = (v8f){};

#pragma unroll 4
    for (int k = 0; k < 128; k += 4) {
        int kb = k + kg * 2;
        v2f a = *(const v2f*)(arow + kb);
        if (HASBIAS) {
            a.x = fmaxf(a.x + Blds[kb], 0.0f);
            a.y = fmaxf(a.y + Blds[kb + 1], 0.0f);
        }
        int pbase = ((kb >> 1) << 7);
#pragma unroll
        for (int j = 0; j < 8; ++j) {
            int col = j * 16 + m;
            v2f b = *(const v2f*)&Wlds[(pbase + col) * 2];
            acc[j] = __builtin_amdgcn_wmma_f32_16x16x4_f32(
                false, a, false, b, (short)0, acc[j], false, false);
        }
    }

    int rOff = kg * 8;
#pragma unroll
    for (int j = 0; j < 8; ++j) {
#pragma unroll
        for (int r = 0; r < 8; ++r) {
            int orow = rowBase + r + rOff;
            if (orow < n) out[(size_t)orow * HID + j * 16 + m] = acc[j][r];
        }
    }
}

// ---------------------------------------------------------------------------
// Fused head: h2 = relu(agg2 + gcn2_b); h = [h2, action];
// h3 = relu(h @ fc1_w + fc1_b); q = h3 @ fc2_w + fc2_b.
// K=144 (128 agg2-path + 16 action), then wave32 butterfly for the fc2 dot.
// ---------------------------------------------------------------------------
__global__ __launch_bounds__(256) void wmma_fc_final_k(
    const float* __restrict__ agg2, const float* __restrict__ bias2,
    const float* __restrict__ action,
    const float* __restrict__ fc1w, const float* __restrict__ fc1b,
    const float* __restrict__ fc2w, const float* __restrict__ fc2b,
    float* __restrict__ q, int n) {
    __shared__ float Wlds[144 * 128];
    __shared__ float B2[128], B1[128], W2v[128];
    int tid = threadIdx.x;
    load_w_paired(fc1w, Wlds, 144, tid, 256);
    if (tid < 128) { B2[tid] = bias2[tid]; B1[tid] = fc1b[tid]; W2v[tid] = fc2w[tid]; }
    __syncthreads();

    int lane = tid & 31;
    int wave = tid >> 5;
    int m  = lane & 15;
    int kg = lane >> 4;
    int rowBase = blockIdx.x * 128 + wave * 16;
    int row = rowBase + m;
    int rowC = row < n ? row : n - 1;
    const float* arow = agg2 + (size_t)rowC * HID;
    const float* act  = action + (size_t)rowC * 16;

    v8f acc[8];
#pragma unroll
    for (int j = 0; j < 8; ++j) acc[j] = (v8f){};

    // K = 0..127 : relu(agg2 + b2) path
#pragma unroll 4
    for (int k = 0; k < 128; k += 4) {
        int kb = k + kg * 2;
        v2f a = *(const v2f*)(arow + kb);
        a.x = fmaxf(a.x + B2[kb], 0.0f);
        a.y = fmaxf(a.y + B2[kb + 1], 0.0f);
        int pbase = ((kb >> 1) << 7);
#pragma unroll
        for (int j = 0; j < 8; ++j) {
            int col = j * 16 + m;
            v2f b = *(const v2f*)&Wlds[(pbase + col) * 2];
            acc[j] = __builtin_amdgcn_wmma_f32_16x16x4_f32(
                false, a, false, b, (short)0, acc[j], false, false);
        }
    }
    // K = 128..143 : action path (raw)
#pragma unroll
    for (int k = 128; k < 144; k += 4) {
        int kb = k + kg * 2;
        v2f a = *(const v2f*)(act + (kb - 128));
        int pbase = ((kb >> 1) << 7);
#pragma unroll
        for (int j = 0; j < 8; ++j) {
            int col = j * 16 + m;
            v2f b = *(const v2f*)&Wlds[(pbase + col) * 2];
            acc[j] = __builtin_amdgcn_wmma_f32_16x16x4_f32(
                false, a, false, b, (short)0, acc[j], false, false);
        }
    }

    // fc2: per-lane partial over its 8 columns
    float s[8];
#pragma unroll
    for (int r = 0; r < 8; ++r) s[r] = 0.0f;
#pragma unroll
    for (int j = 0; j < 8; ++j) {
        int col = j * 16 + m;
        float b1 = B1[col], w2 = W2v[col];
#pragma unroll
        for (int r = 0; r < 8; ++r)
            s[r] += fmaxf(acc[j][r] + b1, 0.0f) * w2;
    }
    // butterfly across the 16 column-lanes of each half-wave
#pragma unroll
    for (int mask = 1; mask < 16; mask <<= 1) {
#pragma unroll
        for (int r = 0; r < 8; ++r) s[r] += __shfl_xor(s[r], mask, 32);
    }

    if (m == 0) {
        float c2 = fc2b[0];
        int rOff = kg * 8;
#pragma unroll
        for (int r = 0; r < 8; ++r) {
            int orow = rowBase + r + rOff;
            if (orow < n) q[orow] = s[r] + c2;
        }
    }
}

// ---------------------------------------------------------------------------
extern "C" void kernel_launch(void* const* d_in, const int* in_sizes, int n_in,
                              void* d_out, int out_size, void* d_ws, size_t ws_size,
                              hipStream_t stream) {
    const float* x      = (const float*)d_in[0];
    const int*   ei     = (const int*)d_in[1];
    const float* action = (const float*)d_in[2];
    const float* g1w = (const float*)d_in[3];
    const float* g1b = (const float*)d_in[4];
    const float* g2w = (const float*)d_in[5];
    const float* g2b = (const float*)d_in[6];
    const float* f1w = (const float*)d_in[7];
    const float* f1b = (const float*)d_in[8];
    const float* f2w = (const float*)d_in[9];
    const float* f2b = (const float*)d_in[10];
    float* q = (float*)d_out;

    int N = in_sizes[0] / STATE_DIM;
    int E = in_sizes[1] / 2;
    const int* srcIdx = ei;
    const int* dstIdx = ei + E;

    char* ws = (char*)d_ws;
    size_t nb = (size_t)N * HID * sizeof(float);
    size_t nvec = (((size_t)N * sizeof(float)) + 255) & ~(size_t)255;
    float* xw   = (float*)ws;
    float* agg  = (float*)(ws + nb);
    float* deg  = (float*)(ws + 2 * nb);
    float* dinv = (float*)(ws + 2 * nb + nvec);

    int nBlk = (N + 255) / 256;
    int eBlk = (E + 255) / 256;
    deg_init_k<<<nBlk, 256, 0, stream>>>(deg, N);
    deg_edge_k<<<eBlk, 256, 0, stream>>>(dstIdx, deg, E);
    dinv_k<<<nBlk, 256, 0, stream>>>(deg, dinv, N);

    int gemmBlk = (N + 127) / 128;
    long long slThreads = (long long)N * 32;
    long long esThreads = (long long)E * 32;
    int slBlk = (int)((slThreads + 255) / 256);
    int esBlk = (int)((esThreads + 255) / 256);

    // Layer 1: xw = x @ W1 ; agg = norm-aggregate(xw)
    wmma_gemm128_k<false><<<gemmBlk, 256, 0, stream>>>(x, g1w, nullptr, xw, N);
    selfloop_k<<<slBlk, 256, 0, stream>>>(xw, dinv, agg, N);
    edge_scatter_k<<<esBlk, 256, 0, stream>>>(xw, dinv, srcIdx, dstIdx, agg, E);

    // Layer 2: xw = relu(agg + b1) @ W2 ; agg = norm-aggregate(xw)
    wmma_gemm128_k<true><<<gemmBlk, 256, 0, stream>>>(agg, g2w, g1b, xw, N);
    selfloop_k<<<slBlk, 256, 0, stream>>>(xw, dinv, agg, N);
    edge_scatter_k<<<esBlk, 256, 0, stream>>>(xw, dinv, srcIdx, dstIdx, agg, E);

    // Head: relu(agg + b2) -> concat action -> fc1 -> relu -> fc2
    wmma_fc_final_k<<<gemmBlk, 256, 0, stream>>>(agg, g2b, action, f1w, f1b, f2w, f2b, q, N);
}